// ROIAlign_37958920962101
// MI455X (gfx1250) — compile-verified
//
#include <hip/hip_runtime.h>

// ROIAlign / tf.crop_and_resize, bilinear, extrapolation_value = 0.
// fmap: (8,128,128,256) f32, boxes: (N,4) f32 [y1,x1,y2,x2], box_idx: (N,) i32
// out:  (N,7,7,256) f32
//
// Bandwidth-bound (AI ~ 0.7 FLOP/B). Strategy for MI455X:
//  - fully coalesced b128 gathers/stores (channel dim contiguous, 64 lanes x 16B)
//  - NT-hinted output stores so the 100MB output stream does not evict the
//    134MB feature map from the 192MB L2 (loads stay RT -> L2 resident)
//  - one workgroup per box so WGP$ captures duplicate-corner reuse
//  - pixel loop unrolled x2: 8 outstanding b128 loads per wave for MLP

typedef float v4f __attribute__((ext_vector_type(4)));

#define FM_H 128
#define FM_W 128
#define FM_C 256
#define POOL 7

__global__ __launch_bounds__(256)
void roialign_gfx1250_kernel(const float* __restrict__ fmap,
                             const float* __restrict__ boxes,
                             const int*   __restrict__ box_idx,
                             float*       __restrict__ out,
                             int n_boxes)
{
    const int b = blockIdx.x;
    if (b >= n_boxes) return;

    // Wave-uniform box parameters (compiler scalarizes these to s_load)
    const float y1 = boxes[b * 4 + 0];
    const float x1 = boxes[b * 4 + 1];
    const float y2 = boxes[b * 4 + 2];
    const float x2 = boxes[b * 4 + 3];
    const int   bi = box_idx[b];

    // True division by 6.0f (once per block) keeps bit-faithfulness to the ref.
    const float h_scale = (y2 - y1) * (float)(FM_H - 1) / (float)(POOL - 1);
    const float w_scale = (x2 - x1) * (float)(FM_W - 1) / (float)(POOL - 1);
    const float ybase   = y1 * (float)(FM_H - 1);
    const float xbase   = x1 * (float)(FM_W - 1);

    const int tid   = threadIdx.x;
    const int group = tid >> 6;   // 4 groups of 64 lanes (2 waves each)
    const int lane  = tid & 63;   // 64 lanes x float4 = 256 channels

    const float* fm    = fmap + (size_t)bi * (FM_H * FM_W * FM_C);
    float*       obase = out  + (size_t)b  * (POOL * POOL * FM_C);

    const v4f zero = {0.0f, 0.0f, 0.0f, 0.0f};

    // Independent iterations; unroll x2 to double outstanding b128 loads.
#pragma unroll 2
    for (int pix = group; pix < POOL * POOL; pix += 4) {
        const int py = pix / POOL;
        const int px = pix - py * POOL;

        const float in_y = ybase + (float)py * h_scale;
        const float in_x = xbase + (float)px * w_scale;
        const float y0f  = floorf(in_y);
        const float x0f  = floorf(in_x);
        const float ly   = in_y - y0f;
        const float lx   = in_x - x0f;

        const int y0 = (int)fminf(fmaxf(y0f,        0.0f), (float)(FM_H - 1));
        const int yb = (int)fminf(fmaxf(y0f + 1.0f, 0.0f), (float)(FM_H - 1));
        const int x0 = (int)fminf(fmaxf(x0f,        0.0f), (float)(FM_W - 1));
        const int xr = (int)fminf(fmaxf(x0f + 1.0f, 0.0f), (float)(FM_W - 1));

        const bool valid = (in_y >= 0.0f) && (in_y <= (float)(FM_H - 1)) &&
                           (in_x >= 0.0f) && (in_x <= (float)(FM_W - 1));

        // Coalesced 1KB channel-row reads per corner (64 lanes x b128)
        const v4f* r00 = (const v4f*)(fm + ((size_t)y0 * FM_W + x0) * FM_C) + lane;
        const v4f* r01 = (const v4f*)(fm + ((size_t)y0 * FM_W + xr) * FM_C) + lane;
        const v4f* r10 = (const v4f*)(fm + ((size_t)yb * FM_W + x0) * FM_C) + lane;
        const v4f* r11 = (const v4f*)(fm + ((size_t)yb * FM_W + xr) * FM_C) + lane;

        const v4f v00 = *r00;
        const v4f v01 = *r01;
        const v4f v10 = *r10;
        const v4f v11 = *r11;

        // Faithful to reference FP ordering: a*(1-l) + b*l
        const v4f top = v00 * (1.0f - lx) + v01 * lx;
        const v4f bot = v10 * (1.0f - lx) + v11 * lx;
        v4f       res = top * (1.0f - ly) + bot * ly;
        if (!valid) res = zero;

        // Non-temporal store: keep L2 (192MB) for the 134MB feature map;
        // lowers to global_store_b128 with NT temporal hint on gfx1250.
        __builtin_nontemporal_store(res, (v4f*)(obase + (size_t)pix * FM_C) + lane);
    }
}

extern "C" void kernel_launch(void* const* d_in, const int* in_sizes, int n_in,
                              void* d_out, int out_size, void* d_ws, size_t ws_size,
                              hipStream_t stream) {
    const float* fmap    = (const float*)d_in[0];
    const float* boxes   = (const float*)d_in[1];
    const int*   box_idx = (const int*)d_in[2];
    float*       out     = (float*)d_out;

    const int n_boxes = in_sizes[2];  // 2000

    roialign_gfx1250_kernel<<<n_boxes, 256, 0, stream>>>(fmap, boxes, box_idx, out, n_boxes);
}